// GNNModel_13735305412781
// MI455X (gfx1250) — compile-verified
//
#include <hip/hip_runtime.h>

#define N_NODES 100000
#define N_EDGES 1000000
#define F 64
#define NTILES (N_NODES / 16)   // 6250, exact

typedef __attribute__((ext_vector_type(2))) float v2f;
typedef __attribute__((ext_vector_type(8))) float v8f;

// ---------- degree / normalization precompute ----------

__global__ void deg_init_kernel(float* deg, int n) {
    int i = blockIdx.x * blockDim.x + threadIdx.x;
    if (i < n) deg[i] = 1.0f;              // self-loop contributes 1
}

__global__ void deg_accum_kernel(const long long* __restrict__ ei, float* deg, int e_cnt) {
    int e = blockIdx.x * blockDim.x + threadIdx.x;
    if (e < e_cnt) {
        int d = (int)ei[N_EDGES + e];      // row 1 = dst
        atomicAdd(&deg[d], 1.0f);
    }
}

__global__ void dinv_kernel(float* deg, int n) {
    int i = blockIdx.x * blockDim.x + threadIdx.x;
    if (i < n) deg[i] = rsqrtf(deg[i]);    // deg >= 1 always (self-loop)
}

__global__ void norm_kernel(const long long* __restrict__ ei,
                            const float* __restrict__ dinv,
                            float* __restrict__ norm, int e_cnt) {
    int e = blockIdx.x * blockDim.x + threadIdx.x;
    if (e < e_cnt) {
        int s = (int)ei[e];
        int d = (int)ei[N_EDGES + e];
        norm[e] = dinv[s] * dinv[d];
    }
}

// ---------- dense: out[N,64] = in[N,64] @ W[64,64] via V_WMMA_F32_16X16X4_F32 ----------
// One wave32 per 16-row tile. A fragment (16x4 f32): lane L holds row L%16,
// K = 4*kt + 2*(L/16) + {0,1}.  B fragment (4x16): lane L holds col L%16, same K pair.
// C/D (16x16 f32, 8 VGPRs): M = v + 8*(L>=16), N = L%16.

__global__ void __launch_bounds__(256)
gemm64_wmma_kernel(const float* __restrict__ in,
                   const float* __restrict__ W,
                   float* __restrict__ out,
                   int num_tiles) {
    int wave = blockIdx.x * (blockDim.x >> 5) + (threadIdx.x >> 5);
    if (wave >= num_tiles) return;         // wave-uniform; active waves keep EXEC all-1s
    int lane = threadIdx.x & 31;
    int r    = lane & 15;
    int kh   = (lane >> 4) << 1;           // 0 or 2

    // Load A fragments for all 16 K-tiles (resident: 32 VGPRs)
    const float* ap = in + (size_t)(wave * 16 + r) * F;
    v2f a[16];
#pragma unroll
    for (int kt = 0; kt < 16; ++kt) {
        a[kt].x = ap[4 * kt + kh];
        a[kt].y = ap[4 * kt + kh + 1];
    }

    int mbase = wave * 16 + ((lane >> 4) << 3);   // +8 row offset for upper half-wave
#pragma unroll
    for (int nt = 0; nt < 4; ++nt) {
        int col = nt * 16 + r;
        v8f acc = {};
#pragma unroll
        for (int kt = 0; kt < 16; ++kt) {
            v2f b;
            b.x = W[(4 * kt + kh) * F + col];
            b.y = W[(4 * kt + kh + 1) * F + col];
            acc = __builtin_amdgcn_wmma_f32_16x16x4_f32(
                /*neg_a=*/false, a[kt], /*neg_b=*/false, b,
                /*c_mod=*/(short)0, acc, /*reuse_a=*/false, /*reuse_b=*/false);
        }
        float* op = out + col;
#pragma unroll
        for (int v = 0; v < 8; ++v)
            op[(size_t)(mbase + v) * F] = acc[v];
    }
}

// ---------- sparse aggregation ----------
// init: self-loop message + bias (also overwrites poisoned output buffer)
__global__ void agg_init_kernel(const float* __restrict__ h,
                                const float* __restrict__ dinv,
                                const float* __restrict__ bias,
                                float* __restrict__ out, int total) {
    int gid = blockIdx.x * blockDim.x + threadIdx.x;
    if (gid < total) {
        int i = gid >> 6;
        int f = gid & 63;
        float di = dinv[i];
        out[gid] = h[gid] * di * di + bias[f];
    }
}

// edges: 64 consecutive threads handle one edge -> fully coalesced 256B gather
// + coalesced f32 atomic scatter. Edge scalars are wave-uniform (scalarized).
__global__ void agg_edges_kernel(const long long* __restrict__ ei,
                                 const float* __restrict__ norm,
                                 const float* __restrict__ h,
                                 float* __restrict__ out, int total) {
    int gid = blockIdx.x * blockDim.x + threadIdx.x;
    if (gid < total) {
        int e = gid >> 6;
        int f = gid & 63;
        int s = (int)ei[e];
        int d = (int)ei[N_EDGES + e];
        atomicAdd(&out[(size_t)d * F + f], h[(size_t)s * F + f] * norm[e]);
    }
}

__global__ void relu_kernel(float* buf, int total) {
    int gid = blockIdx.x * blockDim.x + threadIdx.x;
    if (gid < total) buf[gid] = fmaxf(buf[gid], 0.0f);
}

// ---------- launch ----------

extern "C" void kernel_launch(void* const* d_in, const int* in_sizes, int n_in,
                              void* d_out, int out_size, void* d_ws, size_t ws_size,
                              hipStream_t stream) {
    const float*     x  = (const float*)d_in[0];
    const long long* ei = (const long long*)d_in[1];   // int64 in setup_inputs
    const float*     W1 = (const float*)d_in[2];
    const float*     b1 = (const float*)d_in[3];
    const float*     W2 = (const float*)d_in[4];
    const float*     b2 = (const float*)d_in[5];
    float*           out = (float*)d_out;

    float* ws   = (float*)d_ws;
    float* dinv = ws;                               // N floats (deg -> dinv in place)
    float* norm = dinv + N_NODES;                   // E floats
    float* bufA = norm + N_EDGES;                   // N*F floats (gemm out)
    float* bufB = bufA + (size_t)N_NODES * F;       // N*F floats (layer-1 activations)

    const int TB = 256;
    int gN   = (N_NODES + TB - 1) / TB;             // per-node
    int gE   = (N_EDGES + TB - 1) / TB;             // per-edge
    int gNF  = (N_NODES * F + TB - 1) / TB;         // per node-feature
    int gEF  = (N_EDGES * F + TB - 1) / TB;         // per edge-feature
    int gGemm = (NTILES * 32 + TB - 1) / TB;        // 8 waves/block, 1 tile/wave

    // normalization coefficients (shared by both layers)
    deg_init_kernel <<<gN, TB, 0, stream>>>(dinv, N_NODES);
    deg_accum_kernel<<<gE, TB, 0, stream>>>(ei, dinv, N_EDGES);
    dinv_kernel     <<<gN, TB, 0, stream>>>(dinv, N_NODES);
    norm_kernel     <<<gE, TB, 0, stream>>>(ei, dinv, norm, N_EDGES);

    // layer 1: h = x@W1 ; aggregate ; +b1 ; relu
    gemm64_wmma_kernel<<<gGemm, TB, 0, stream>>>(x, W1, bufA, NTILES);
    agg_init_kernel  <<<gNF, TB, 0, stream>>>(bufA, dinv, b1, bufB, N_NODES * F);
    agg_edges_kernel <<<gEF, TB, 0, stream>>>(ei, norm, bufA, bufB, N_EDGES * F);
    relu_kernel      <<<gNF, TB, 0, stream>>>(bufB, N_NODES * F);

    // layer 2: z = agg(h@W2) + b2  -> d_out
    gemm64_wmma_kernel<<<gGemm, TB, 0, stream>>>(bufB, W2, bufA, NTILES);
    agg_init_kernel  <<<gNF, TB, 0, stream>>>(bufA, dinv, b2, out, N_NODES * F);
    agg_edges_kernel <<<gEF, TB, 0, stream>>>(ei, norm, bufA, out, N_EDGES * F);
}